// GCN_62474594288071
// MI455X (gfx1250) — compile-verified
//
#include <hip/hip_runtime.h>
#include <cmath>

typedef __attribute__((ext_vector_type(2))) float v2f;
typedef __attribute__((ext_vector_type(8))) float v8f;

// ---------------------------------------------------------------------------
// Copy initial h [N,64] into out[:, 0:64] (out row stride = 256 floats).
// float4-vectorized: N*16 threads.
// ---------------------------------------------------------------------------
__global__ void gcn_copy_h_kernel(const float* __restrict__ h,
                                  float* __restrict__ out, int n_nodes) {
  int i = blockIdx.x * blockDim.x + threadIdx.x;
  if (i >= n_nodes * 16) return;
  int r  = i >> 4;
  int c4 = i & 15;
  float4 v = *((const float4*)(h + (size_t)r * 64) + c4);
  *((float4*)(out + (size_t)r * 256) + c4) = v;
}

// ---------------------------------------------------------------------------
// Zero the aggregation buffer (float4-vectorized).
// ---------------------------------------------------------------------------
__global__ void gcn_zero_kernel(float4* __restrict__ p, int n4) {
  int i = blockIdx.x * blockDim.x + threadIdx.x;
  if (i < n4) p[i] = make_float4(0.f, 0.f, 0.f, 0.f);
}

// ---------------------------------------------------------------------------
// Edge gather-scale-scatter: one wave per edge. Each lane handles 2 features
// (32 lanes x float2 = 64 feats). Edge metadata read via readfirstlane so the
// compiler emits scalar loads. Scatter uses native f32 global atomics.
// hprev has row stride 256 (it is a 64-wide column slice of the output).
// ---------------------------------------------------------------------------
__global__ void gcn_edge_scatter_kernel(const float* __restrict__ hprev,
                                        const float* __restrict__ ew,
                                        const int* __restrict__ src,
                                        const int* __restrict__ dst,
                                        float* __restrict__ agg,
                                        int n_edges) {
  int wave = (blockIdx.x * blockDim.x + threadIdx.x) >> 5;
  int lane = threadIdx.x & 31;
  int e = __builtin_amdgcn_readfirstlane(wave);
  if (e >= n_edges) return;
  int   s = src[e];
  int   d = dst[e];
  float w = ew[e];
  float2 hv = *(const float2*)(hprev + (size_t)s * 256 + lane * 2);
  float* p = agg + (size_t)d * 64 + lane * 2;
  unsafeAtomicAdd(p,     hv.x * w);
  unsafeAtomicAdd(p + 1, hv.y * w);
}

// ---------------------------------------------------------------------------
// out[row, col] = tanh?( sum_k A[row,k] * W[k,col] + bias[col] )
// A: [n_rows, 64] (lda=64), W: [64,64] row-major, out: row stride 256
// (pre-offset to the layer's 64-column slice of the concat output).
//
// 8 waves/block; wave w -> 16x16 tile at (blockIdx.x*32 + (w>>2)*16, (w&3)*16).
// K-loop: 16 x V_WMMA_F32_16X16X4_F32.
//   A 16x4 f32 layout: lane l<16 holds (M=l, K=k0..k0+1) in v[0..1];
//                      lane l>=16 holds (M=l-16, K=k0+2..k0+3)  -> float2 load.
//   B 4x16 f32 layout: VGPR0/1 = K pair; lanes 0-15 K=k0..k0+1 at N=col0+l,
//                      lanes 16-31 K=k0+2..k0+3.
//   C/D 16x16 f32: VGPR v -> row v (lanes 0-15) / row v+8 (lanes 16-31).
// ---------------------------------------------------------------------------
__global__ void gcn_gemm_wmma_kernel(const float* __restrict__ A,
                                     const float* __restrict__ W,
                                     const float* __restrict__ bias,
                                     float* __restrict__ out,
                                     int n_rows, int do_tanh) {
  int wave = threadIdx.x >> 5;
  int lane = threadIdx.x & 31;
  int row0 = blockIdx.x * 32 + (wave >> 2) * 16;
  int col0 = (wave & 3) * 16;
  if (row0 >= n_rows) return;            // wave-uniform: EXEC stays all-1s

  int lh = lane >> 4;                    // lane half: 0 or 1
  int lm = lane & 15;
  const float* arow = A + (size_t)(row0 + lm) * 64 + 2 * lh;
  const float* wcol = W + (size_t)(2 * lh) * 64 + col0 + lm;

  v8f acc = {};
#pragma unroll
  for (int k0 = 0; k0 < 64; k0 += 4) {
    v2f a = *(const v2f*)(arow + k0);
    v2f b;
    b.x = wcol[(size_t)k0 * 64];         // W[k0 + 2*lh    ][col0+lm]
    b.y = wcol[(size_t)k0 * 64 + 64];    // W[k0 + 2*lh + 1][col0+lm]
    acc = __builtin_amdgcn_wmma_f32_16x16x4_f32(
        /*neg_a=*/false, a, /*neg_b=*/false, b,
        /*c_mod=*/(short)0, acc, /*reuse_a=*/false, /*reuse_b=*/false);
  }

  float bv = bias[col0 + lm];
  float* orow = out + (size_t)(row0 + 8 * lh) * 256 + col0 + lm;
#pragma unroll
  for (int v = 0; v < 8; ++v) {
    float val = acc[v] + bv;
    if (do_tanh) val = tanhf(val);
    orow[(size_t)v * 256] = val;
  }
}

// ---------------------------------------------------------------------------
extern "C" void kernel_launch(void* const* d_in, const int* in_sizes, int n_in,
                              void* d_out, int out_size, void* d_ws, size_t ws_size,
                              hipStream_t stream) {
  const float* h  = (const float*)d_in[0];   // [N,64]
  const float* ew = (const float*)d_in[1];   // [E]
  const float* Ws = (const float*)d_in[2];   // [3,64,64]
  const float* bs = (const float*)d_in[3];   // [3,64]
  const int*  src = (const int*)d_in[4];     // [E]
  const int*  dst = (const int*)d_in[5];     // [E]
  float* out = (float*)d_out;                // [N,256]
  float* agg = (float*)d_ws;                 // [N,64] scratch

  const int N = in_sizes[0] / 64;
  const int E = in_sizes[1];
  const int L = 3;

  // h -> out[:, 0:64]
  {
    int t = N * 16;
    gcn_copy_h_kernel<<<(t + 255) / 256, 256, 0, stream>>>(h, out, N);
  }

  for (int layer = 0; layer < L; ++layer) {
    // agg = 0
    {
      int n4 = N * 16;
      gcn_zero_kernel<<<(n4 + 255) / 256, 256, 0, stream>>>((float4*)agg, n4);
    }
    // agg[dst] += h_prev[src] * ew   (h_prev = out columns [layer*64, +64))
    {
      long threads = (long)E * 32;
      int blocks = (int)((threads + 255) / 256);
      gcn_edge_scatter_kernel<<<blocks, 256, 0, stream>>>(
          out + layer * 64, ew, src, dst, agg, E);
    }
    // out[:, (layer+1)*64 : +64] = tanh?(agg @ W[layer] + b[layer])
    {
      int blocks = (N + 31) / 32;
      gcn_gemm_wmma_kernel<<<blocks, 256, 0, stream>>>(
          agg, Ws + layer * 64 * 64, bs + layer * 64,
          out + (layer + 1) * 64, N, (layer < L - 1) ? 1 : 0);
    }
  }
}